// FoamEnsemble_39977555591280
// MI455X (gfx1250) — compile-verified
//
#include <hip/hip_runtime.h>
#include <math.h>

// Problem constants
#define V 50257
#define DD 128
#define KF 3
#define NB 8
#define TT 64
#define NDIR 24   // K*NB

// d_out layout (floats), in reference return order
#define OFF_RHO   ((long)TT * V)                         // 3,216,448
#define OFF_S     (OFF_RHO + (long)TT * DD * DD)         // 4,265,024
#define OFF_H     (OFF_S + TT)
#define OFF_F     (OFF_H + TT)
#define OFF_PF    (OFF_F + TT)                           // per_foam_probs

// workspace layout (floats)
#define WS_PSI   0L
#define WS_G     (WS_PSI + NDIR * DD)        // 3072
#define WS_WALL  (WS_G + NDIR * NDIR)        // +576
#define WS_PROJ  (WS_WALL + (long)TT * NDIR) // +1536

typedef float v2f __attribute__((ext_vector_type(2)));
typedef float v8f __attribute__((ext_vector_type(8)));

__device__ __forceinline__ float waveSum(float v) {
  #pragma unroll
  for (int o = 16; o > 0; o >>= 1) v += __shfl_xor(v, o, 32);
  return v;
}
__device__ __forceinline__ float waveMax(float v) {
  #pragma unroll
  for (int o = 16; o > 0; o >>= 1) v = fmaxf(v, __shfl_xor(v, o, 32));
  return v;
}

// ---------------- psi = bubbles / (||bubbles|| + 1e-10) ----------------
__global__ void k_psi(const float* __restrict__ bubbles, float* __restrict__ psi) {
  int j = blockIdx.x;            // 0..23
  int tid = threadIdx.x;         // 0..127
  __shared__ float part[4];
  __shared__ float nrm;
  float v = bubbles[j * DD + tid];
  float s = waveSum(v * v);
  if ((tid & 31) == 0) part[tid >> 5] = s;
  __syncthreads();
  if (tid == 0) nrm = sqrtf(part[0] + part[1] + part[2] + part[3]) + 1e-10f;
  __syncthreads();
  psi[j * DD + tid] = v / nrm;
}

// ---------------- G = Psi Psi^T (24x24, fixed) ----------------
__global__ void k_gram(const float* __restrict__ psi, float* __restrict__ G) {
  int gid = blockIdx.x * 256 + threadIdx.x;
  if (gid >= NDIR * NDIR) return;
  int i = gid / NDIR, j = gid % NDIR;
  float s = 0.f;
  for (int d = 0; d < DD; ++d) s += psi[i * DD + d] * psi[j * DD + d];
  G[gid] = s;
}

// ---------------- sequential scan -> w_all[T][24] ----------------
__global__ void k_scan(const int* __restrict__ tokens, const float* __restrict__ E,
                       const float* __restrict__ bubbles,
                       const float* __restrict__ decay_base_p,
                       const float* __restrict__ sens_p,
                       float* __restrict__ w_all) {
  __shared__ float mem[NDIR * DD];     // [24][128]
  __shared__ float xbuf[DD];
  __shared__ float mm[KF * DD];
  __shared__ float xm[KF * DD];
  __shared__ float memnorm2[KF], dotmx[KF], xnorm2_s;
  __shared__ float decay_s[KF];
  __shared__ float scores[NDIR], wbuf[NDIR];
  int tid = threadIdx.x;               // 256 threads = 8 waves
  int lane = tid & 31, wv = tid >> 5;
  for (int i = tid; i < NDIR * DD; i += 256) mem[i] = 0.f;
  __syncthreads();
  float db = decay_base_p[0];
  float sens = fabsf(sens_p[0]);
  const float inv_sqrt_d = 0.08838834764831845f;   // 1/sqrt(128)
  for (int t = 0; t < TT; ++t) {
    int tok = tokens[t];
    if (tid < DD) xbuf[tid] = E[(long)tok * DD + tid];
    __syncthreads();
    if (tid < KF * DD) {
      int k = tid >> 7, d = tid & 127;
      float s = 0.f;
      #pragma unroll
      for (int n = 0; n < NB; ++n) s += mem[(k * NB + n) * DD + d];
      mm[tid] = s * (1.f / NB);
    }
    __syncthreads();
    if (wv < KF) {
      float an = 0.f, ad = 0.f;
      for (int d = lane; d < DD; d += 32) { float m = mm[wv * DD + d]; an += m * m; ad += m * xbuf[d]; }
      an = waveSum(an); ad = waveSum(ad);
      if (lane == 0) { memnorm2[wv] = an; dotmx[wv] = ad; }
    } else if (wv == KF) {
      float ax = 0.f;
      for (int d = lane; d < DD; d += 32) { float x = xbuf[d]; ax += x * x; }
      ax = waveSum(ax);
      if (lane == 0) xnorm2_s = ax;
    }
    __syncthreads();
    if (tid < KF) {
      float mn = sqrtf(memnorm2[tid]) + 1e-10f;
      float xn = sqrtf(xnorm2_s) + 1e-10f;
      float cosv = dotmx[tid] / (xn * mn);
      float nov = (mn > 1e-8f) ? (1.f - cosv) : 1.f;
      float z = db - sens * nov;
      decay_s[tid] = 1.f / (1.f + expf(-z));
    }
    __syncthreads();
    if (tid < KF * DD) {
      int k = tid >> 7, d = tid & 127;
      xm[tid] = xbuf[d] + decay_s[k] * mm[tid];
    }
    __syncthreads();
    // 24 score dots: wave wv handles j = wv + 8*c
    #pragma unroll
    for (int c = 0; c < 3; ++c) {
      int j = wv + 8 * c;
      int k = j >> 3;
      float acc = 0.f;
      for (int d = lane; d < DD; d += 32) acc += bubbles[j * DD + d] * xm[k * DD + d];
      acc = waveSum(acc);
      if (lane == 0) scores[j] = 2.0f * acc * inv_sqrt_d;
    }
    __syncthreads();
    if (tid < KF) {
      float mx = -1e30f;
      #pragma unroll
      for (int n = 0; n < NB; ++n) mx = fmaxf(mx, scores[tid * NB + n]);
      float e[NB]; float s = 0.f;
      #pragma unroll
      for (int n = 0; n < NB; ++n) { e[n] = expf(scores[tid * NB + n] - mx); s += e[n]; }
      #pragma unroll
      for (int n = 0; n < NB; ++n) wbuf[tid * NB + n] = e[n] / s;
    }
    __syncthreads();
    if (tid < NDIR) w_all[t * NDIR + tid] = wbuf[tid];
    for (int i = tid; i < NDIR * DD; i += 256) {
      int j = i >> 7, d = i & 127, k = j >> 3;
      float dcy = decay_s[k], wj = wbuf[j];
      float eq = wj * xm[k * DD + d] + (1.f - wj) * bubbles[i];
      mem[i] = dcy * mem[i] + (1.f - dcy) * eq;
    }
    __syncthreads();
  }
}

// ---------------- proj = E @ Psi^T via V_WMMA_F32_16X16X4_F32 ----------------
__global__ void k_proj(const float* __restrict__ E, const float* __restrict__ psi,
                       float* __restrict__ proj) {
  __shared__ float psiT[DD][32];       // [k][j], zero-padded j=24..31
  int tid = threadIdx.x;               // 128 threads = 4 waves
  {
    int d = tid;
    #pragma unroll
    for (int j = 0; j < 32; ++j)
      psiT[d][j] = (j < NDIR) ? psi[j * DD + d] : 0.f;
  }
  __syncthreads();
  int lane = tid & 31, wv = tid >> 5;
  long tile = (long)blockIdx.x * 4 + wv;
  long row0 = tile * 16;
  if (row0 >= V) return;               // wave-uniform exit; EXEC stays all-ones for WMMA
  int m16 = lane & 15, hi = lane >> 4;
  long row = row0 + m16; if (row > V - 1) row = V - 1;   // clamp (stores are guarded)
  const float* erow = E + row * DD;
  __builtin_prefetch(erow, 0, 0);
  v8f acc0 = {0.f,0.f,0.f,0.f,0.f,0.f,0.f,0.f};
  v8f acc1 = {0.f,0.f,0.f,0.f,0.f,0.f,0.f,0.f};
  #pragma unroll
  for (int k = 0; k < DD; k += 4) {
    int kk = k + 2 * hi;
    // A 16x4 f32 layout: lanes 0-15 hold K={0,1}, lanes 16-31 hold K={2,3}
    v2f a; a.x = erow[kk]; a.y = erow[kk + 1];
    // B 4x16 f32 layout: VGPR0 = rows K={0,2}, VGPR1 = rows K={1,3}
    v2f b0; b0.x = psiT[kk][m16];      b0.y = psiT[kk + 1][m16];
    v2f b1; b1.x = psiT[kk][16 + m16]; b1.y = psiT[kk + 1][16 + m16];
    acc0 = __builtin_amdgcn_wmma_f32_16x16x4_f32(false, a, false, b0, (short)0, acc0, false, false);
    acc1 = __builtin_amdgcn_wmma_f32_16x16x4_f32(false, a, false, b1, (short)0, acc1, false, false);
  }
  // D layout: lane&15 = N, VGPR r -> M = r + 8*(lane>>4)
  #pragma unroll
  for (int r = 0; r < 8; ++r) {
    long m = row0 + r + 8 * hi;
    if (m < V) {
      proj[m * NDIR + m16] = acc0[r];
      if (m16 < NDIR - 16) proj[m * NDIR + 16 + m16] = acc1[r];
    }
  }
}

// ---------------- per-step S_rho via 24x24 Jacobi (one wave per t) ----------------
__global__ void k_eigen(const float* __restrict__ G, const float* __restrict__ w_all,
                        float* __restrict__ S_out) {
  __shared__ float M[NDIR][25];
  int t = blockIdx.x;
  int lane = threadIdx.x;              // 32
  const float* w = w_all + t * NDIR;
  for (int idx = lane; idx < NDIR * NDIR; idx += 32) {
    int i = idx / NDIR, j = idx % NDIR;
    float li = w[i] * (1.f / KF), lj = w[j] * (1.f / KF);
    M[i][j] = sqrtf(li * lj) * G[i * NDIR + j];
  }
  __syncthreads();
  for (int sweep = 0; sweep < 10; ++sweep) {
    for (int p = 0; p < NDIR - 1; ++p) {
      for (int q = p + 1; q < NDIR; ++q) {
        float apq = M[p][q];
        if (fabsf(apq) > 1e-14f) {       // wave-uniform
          float app = M[p][p], aqq = M[q][q];
          float tau = (aqq - app) / (2.f * apq);
          float tt = (tau >= 0.f ? 1.f : -1.f) / (fabsf(tau) + sqrtf(1.f + tau * tau));
          float c = 1.f / sqrtf(1.f + tt * tt);
          float s = tt * c;
          int j = lane;
          bool act = (j < NDIR) && (j != p) && (j != q);
          float npj = 0.f, nqj = 0.f;
          if (act) {
            float apj = M[p][j], aqj = M[q][j];
            npj = c * apj - s * aqj;
            nqj = s * apj + c * aqj;
          }
          __syncthreads();
          if (act) { M[p][j] = npj; M[j][p] = npj; M[q][j] = nqj; M[j][q] = nqj; }
          if (lane == 0) {
            M[p][p] = app - tt * apq;
            M[q][q] = aqq + tt * apq;
            M[p][q] = 0.f; M[q][p] = 0.f;
          }
          __syncthreads();
        }
      }
    }
  }
  // entropy: 24 real eigenvalues + (128-24) emulated zeros clipped to 1e-12
  float e = (lane < NDIR) ? fmaxf(M[lane][lane], 1e-12f) : 0.f;
  float sum = waveSum(e) + (float)(DD - NDIR) * 1e-12f;
  float inv = 1.f / sum;
  float term = 0.f;
  if (lane < NDIR) { float pr = e * inv; term = pr * fmaxf(logf(pr), -100.f); }
  float S = waveSum(term);
  float t0 = 1e-12f * inv;
  S += (float)(DD - NDIR) * (t0 * fmaxf(logf(t0), -100.f));
  if (lane == 0) S_out[t] = -S;
}

// ---------------- rho[t] output: sum_j (w_j/K) psi_j psi_j^T ----------------
__global__ void k_rho(const float* __restrict__ psi, const float* __restrict__ w_all,
                      float* __restrict__ out) {
  int t = blockIdx.x >> 7;
  int d = blockIdx.x & 127;
  __shared__ float coef[NDIR];
  int tid = threadIdx.x;               // 128
  if (tid < NDIR) coef[tid] = (w_all[t * NDIR + tid] * (1.f / KF)) * psi[tid * DD + d];
  __syncthreads();
  float acc = 0.f;
  #pragma unroll
  for (int j = 0; j < NDIR; ++j) acc += coef[j] * psi[j * DD + tid];
  out[OFF_RHO + (long)t * DD * DD + d * DD + tid] = acc;
}

// ---------------- logits: per_foam + mean (written as logits; softmax later) ----------------
__global__ void k_logits(const float* __restrict__ proj, const float* __restrict__ w_all,
                         float* __restrict__ out) {
  const int nvb = (V + 255) / 256;     // 197
  int t = blockIdx.x / nvb;
  int vb = blockIdx.x % nvb;
  __shared__ float wsh[NDIR];
  if (threadIdx.x < NDIR) wsh[threadIdx.x] = w_all[t * NDIR + threadIdx.x];
  __syncthreads();
  long v = (long)vb * 256 + threadIdx.x;
  if (v >= V) return;
  const float* pv = proj + v * NDIR;
  float mean = 0.f;
  float* pf = out + OFF_PF;
  #pragma unroll
  for (int k = 0; k < KF; ++k) {
    float s = 0.f;
    #pragma unroll
    for (int n = 0; n < NB; ++n) { float p = pv[k * NB + n]; s += wsh[k * NB + n] * p * p; }
    pf[((long)t * KF + k) * V + v] = s;
    mean += s;
  }
  out[(long)t * V + v] = mean * (1.f / KF);
}

// ---------------- fused softmax (+H, F for token rows) ----------------
__global__ void k_softmax(float* __restrict__ out) {
  int r = blockIdx.x;                  // 0..255: first T*K per-foam rows, then T token rows
  bool token = (r >= TT * KF);
  float* row; int t;
  if (!token) { t = r / KF; row = out + OFF_PF + (long)r * V; }
  else        { t = r - TT * KF; row = out + (long)t * V; }
  int tid = threadIdx.x;               // 256
  __shared__ float red[8];
  __shared__ float s_mx, s_sum;
  float mx = -INFINITY;
  for (long v = tid; v < V; v += 256) mx = fmaxf(mx, row[v]);
  mx = waveMax(mx);
  if ((tid & 31) == 0) red[tid >> 5] = mx;
  __syncthreads();
  if (tid == 0) { float m = red[0]; for (int i = 1; i < 8; ++i) m = fmaxf(m, red[i]); s_mx = m; }
  __syncthreads();
  mx = s_mx;
  float sum = 0.f;
  for (long v = tid; v < V; v += 256) sum += expf(row[v] - mx);
  sum = waveSum(sum);
  if ((tid & 31) == 0) red[tid >> 5] = sum;
  __syncthreads();
  if (tid == 0) { float s = 0.f; for (int i = 0; i < 8; ++i) s += red[i]; s_sum = s; }
  __syncthreads();
  float inv = 1.f / s_sum, lg = logf(s_sum);
  float h = 0.f;
  for (long v = tid; v < V; v += 256) {
    float x = row[v] - mx;
    float p = expf(x) * inv;
    row[v] = p;
    if (token) { float lp = x - lg; h -= p * fmaxf(lp, -100.f); }
  }
  if (token) {
    h = waveSum(h);
    if ((tid & 31) == 0) red[tid >> 5] = h;
    __syncthreads();
    if (tid == 0) {
      float H = 0.f;
      for (int i = 0; i < 8; ++i) H += red[i];
      out[OFF_H + t] = H;
      out[OFF_F + t] = H - out[OFF_S + t];
    }
  }
}

extern "C" void kernel_launch(void* const* d_in, const int* in_sizes, int n_in,
                              void* d_out, int out_size, void* d_ws, size_t ws_size,
                              hipStream_t stream) {
  const int*   tokens      = (const int*)d_in[0];
  const float* E           = (const float*)d_in[1];
  const float* bubbles     = (const float*)d_in[2];
  const float* decay_base  = (const float*)d_in[3];
  const float* sensitivity = (const float*)d_in[4];
  float* out = (float*)d_out;
  float* ws  = (float*)d_ws;
  float* psi   = ws + WS_PSI;
  float* G     = ws + WS_G;
  float* w_all = ws + WS_WALL;
  float* proj  = ws + WS_PROJ;

  k_psi   <<<NDIR, DD, 0, stream>>>(bubbles, psi);
  k_gram  <<<3, 256, 0, stream>>>(psi, G);
  k_scan  <<<1, 256, 0, stream>>>(tokens, E, bubbles, decay_base, sensitivity, w_all);
  {
    int tiles = (V + 15) / 16;             // 3142
    int blocks = (tiles + 3) / 4;          // 786, 4 waves per block
    k_proj<<<blocks, 128, 0, stream>>>(E, psi, proj);
  }
  k_eigen <<<TT, 32, 0, stream>>>(G, w_all, out + OFF_S);
  k_rho   <<<TT * DD, DD, 0, stream>>>(psi, w_all, out);
  k_logits<<<TT * ((V + 255) / 256), 256, 0, stream>>>(proj, w_all, out);
  k_softmax<<<TT * (KF + 1), 256, 0, stream>>>(out);
}